// Sc_encoder_53592601919691
// MI455X (gfx1250) — compile-verified
//
#include <hip/hip_runtime.h>
#include <hip/hip_bf16.h>
#include <math.h>

typedef float v2f __attribute__((ext_vector_type(2)));
typedef float v8f __attribute__((ext_vector_type(8)));

#define D_FEAT 256
#define HEADS 8
#define HIDDEN 32
#define OUT_D 32

// ---------------------------------------------------------------------------
// Kernel 1: feat = x @ W   ([N,256] x [256,256], fp32, WMMA 16x16x4)
// Each wave computes a 16(M) x 64(N) strip: 4 accumulators sharing one A
// fragment per k-step -> 4x less x traffic, 10 B/WMMA instead of 16 B/WMMA.
// N % 16 == 0 (50000 = 3125*16); 256/64 = 4 N-groups.
// ---------------------------------------------------------------------------
__global__ __launch_bounds__(256) void gemm_feat_kernel(
    const float* __restrict__ x, const float* __restrict__ W,
    float* __restrict__ feat, int N, int totalTiles) {
  const int lane = threadIdx.x & 31;
  const int wid  = threadIdx.x >> 5;
  const int tile = blockIdx.x * 8 + wid;            // uniform per wave
  if (tile >= totalTiles) return;                   // wave-uniform exit
  const int nGroups = D_FEAT / 64;                  // 4
  const int mTile  = tile / nGroups;
  const int nGroup = tile % nGroups;

  const int m    = mTile * 16 + (lane & 15);        // A row for this lane
  const int n0   = nGroup * 64 + (lane & 15);       // first B col for this lane
  const int kOff = (lane >> 4) * 2;                 // 0 for lanes 0-15, 2 for 16-31

  v8f acc0 = {0.f, 0.f, 0.f, 0.f, 0.f, 0.f, 0.f, 0.f};
  v8f acc1 = acc0, acc2 = acc0, acc3 = acc0;

  const float* arow = x + (long)m * D_FEAT;
  for (int k = 0; k < D_FEAT; k += 4) {
    const int ka = k + kOff;
    v2f a;
    a.x = arow[ka];
    a.y = arow[ka + 1];
    const float* bp = W + (long)ka * D_FEAT + n0;   // row ka, cols n0..n0+48
    v2f b0, b1, b2, b3;
    b0.x = bp[0];   b0.y = bp[D_FEAT + 0];
    b1.x = bp[16];  b1.y = bp[D_FEAT + 16];
    b2.x = bp[32];  b2.y = bp[D_FEAT + 32];
    b3.x = bp[48];  b3.y = bp[D_FEAT + 48];
    acc0 = __builtin_amdgcn_wmma_f32_16x16x4_f32(false, a, false, b0, (short)0, acc0, false, false);
    acc1 = __builtin_amdgcn_wmma_f32_16x16x4_f32(false, a, false, b1, (short)0, acc1, false, false);
    acc2 = __builtin_amdgcn_wmma_f32_16x16x4_f32(false, a, false, b2, (short)0, acc2, false, false);
    acc3 = __builtin_amdgcn_wmma_f32_16x16x4_f32(false, a, false, b3, (short)0, acc3, false, false);
  }
  // C/D layout: VGPR g -> lanes 0-15: M=g, lanes 16-31: M=g+8; N = lane%16
  const int mBase = mTile * 16 + ((lane >> 4) * 8);
  const int nCol  = nGroup * 64 + (lane & 15);
#pragma unroll
  for (int g = 0; g < 8; ++g) {
    float* frow = feat + (long)(mBase + g) * D_FEAT + nCol;
    frow[0]  = acc0[g];
    frow[16] = acc1[g];
    frow[32] = acc2[g];
    frow[48] = acc3[g];
  }
}

// ---------------------------------------------------------------------------
// Kernel 2: per-node attention halves  el/er[n,h] = sum_d feat[n,h,d]*attn[h,d]
// ---------------------------------------------------------------------------
__global__ __launch_bounds__(256) void node_attn_kernel(
    const float* __restrict__ feat, const float* __restrict__ attn_l,
    const float* __restrict__ attn_r, float* __restrict__ el,
    float* __restrict__ er, long total) {
  long i = (long)blockIdx.x * blockDim.x + threadIdx.x;  // (node, head)
  if (i >= total) return;
  const long node = i >> 3;
  const int h = (int)(i & 7);
  const float* f = feat + node * D_FEAT + h * HIDDEN;
  const float* al = attn_l + h * HIDDEN;
  const float* ar = attn_r + h * HIDDEN;
  float sl = 0.f, sr = 0.f;
#pragma unroll 8
  for (int d = 0; d < HIDDEN; ++d) {
    const float v = f[d];
    sl += v * al[d];
    sr += v * ar[d];
  }
  el[i] = sl;
  er[i] = sr;
}

// ---------------------------------------------------------------------------
// Init workspace: emax = -inf, denom = 0, rst = 0
// ---------------------------------------------------------------------------
__global__ __launch_bounds__(256) void init_ws_kernel(
    float* __restrict__ emax, float* __restrict__ denom,
    float* __restrict__ rst, long nSeg, long nRst) {
  long i = (long)blockIdx.x * blockDim.x + threadIdx.x;
  if (i < nSeg) {
    emax[i]  = -INFINITY;
    denom[i] = 0.f;
  }
  if (i < nRst) rst[i] = 0.f;
}

// float atomic-max via signed/unsigned integer ordering trick
__device__ inline void atomicMaxF(float* addr, float val) {
  if (val >= 0.f) {
    atomicMax((int*)addr, __float_as_int(val));
  } else {
    atomicMin((unsigned int*)addr, __float_as_uint(val));
  }
}

// ---------------------------------------------------------------------------
// Kernel 3: edge logits with LeakyReLU + segment max (atomic)
// ---------------------------------------------------------------------------
__global__ __launch_bounds__(256) void edge_logit_max_kernel(
    const float* __restrict__ el, const float* __restrict__ er,
    const int* __restrict__ src, const int* __restrict__ dst,
    float* __restrict__ e_out, float* __restrict__ emax, long total) {
  long i = (long)blockIdx.x * blockDim.x + threadIdx.x;  // (edge, head)
  if (i >= total) return;
  const long eid = i >> 3;
  const int h = (int)(i & 7);
  const int s = src[eid];
  const int d = dst[eid];
  float e = el[(long)s * HEADS + h] + er[(long)d * HEADS + h];
  e = (e > 0.f) ? e : 0.2f * e;
  e_out[i] = e;
  atomicMaxF(&emax[(long)d * HEADS + h], e);
}

// ---------------------------------------------------------------------------
// Kernel 4: ex = exp(e - emax[dst]); denom[dst] += ex (atomic)
// ---------------------------------------------------------------------------
__global__ __launch_bounds__(256) void edge_expsum_kernel(
    const float* __restrict__ e_in, const int* __restrict__ dst,
    const float* __restrict__ emax, float* __restrict__ ex,
    float* __restrict__ denom, long total) {
  long i = (long)blockIdx.x * blockDim.x + threadIdx.x;  // (edge, head)
  if (i >= total) return;
  const long eid = i >> 3;
  const int h = (int)(i & 7);
  const int d = dst[eid];
  const float v = __expf(e_in[i] - emax[(long)d * HEADS + h]);
  ex[i] = v;
  atomicAdd(&denom[(long)d * HEADS + h], v);
}

// ---------------------------------------------------------------------------
// Kernel 5: rst[dst] += feat[src] * alpha   (4 floats per thread, fp32 atomics;
// whole ~165MB working set is L2-resident on MI455X so gathers/atomics hit L2)
// ---------------------------------------------------------------------------
__global__ __launch_bounds__(256) void aggregate_kernel(
    const float* __restrict__ feat, const float* __restrict__ ex,
    const float* __restrict__ denom, const int* __restrict__ src,
    const int* __restrict__ dst, float* __restrict__ rst, long total) {
  long i = (long)blockIdx.x * blockDim.x + threadIdx.x;  // (edge, chunk-of-4)
  if (i >= total) return;
  const long eid = i >> 6;           // 64 chunks of 4 floats per edge
  const int  c4  = (int)(i & 63);    // chunk index
  const int  c   = c4 * 4;           // feature offset (0..252)
  const int  h   = c >> 5;           // head (32 feats per head, 4|32)
  const int  s = src[eid];
  const int  d = dst[eid];
  const float alpha = ex[eid * HEADS + h] / denom[(long)d * HEADS + h];
  const float4 f = *(const float4*)(feat + (long)s * D_FEAT + c);
  float* r = rst + (long)d * D_FEAT + c;
  atomicAdd(r + 0, f.x * alpha);
  atomicAdd(r + 1, f.y * alpha);
  atomicAdd(r + 2, f.z * alpha);
  atomicAdd(r + 3, f.w * alpha);
}

// ---------------------------------------------------------------------------
// Kernel 6: out = elu(rst + gat_bias) @ lin_W + lin_b   (WMMA; both 16-col
// N-tiles per wave so the fused bias+ELU A fragment is computed once)
// ---------------------------------------------------------------------------
__global__ __launch_bounds__(256) void gemm_out_kernel(
    const float* __restrict__ rst, const float* __restrict__ gat_bias,
    const float* __restrict__ lin_W, const float* __restrict__ lin_b,
    float* __restrict__ out, int N, int totalTiles) {
  const int lane = threadIdx.x & 31;
  const int wid  = threadIdx.x >> 5;
  const int mTile = blockIdx.x * 8 + wid;  // uniform per wave; one M tile/wave
  if (mTile >= totalTiles) return;         // wave-uniform exit

  const int m    = mTile * 16 + (lane & 15);
  const int n0   = (lane & 15);            // cols 0..15 and 16..31
  const int kOff = (lane >> 4) * 2;

  v8f acc0 = {0.f, 0.f, 0.f, 0.f, 0.f, 0.f, 0.f, 0.f};
  v8f acc1 = acc0;
  const float* arow = rst + (long)m * D_FEAT;
  for (int k = 0; k < D_FEAT; k += 4) {
    const int ka = k + kOff;
    float a0 = arow[ka]     + gat_bias[ka];
    float a1 = arow[ka + 1] + gat_bias[ka + 1];
    a0 = (a0 > 0.f) ? a0 : (__expf(a0) - 1.0f);  // ELU (computed once, used twice)
    a1 = (a1 > 0.f) ? a1 : (__expf(a1) - 1.0f);
    v2f a;
    a.x = a0;
    a.y = a1;
    const float* bp = lin_W + (long)ka * OUT_D + n0;
    v2f b0, b1;
    b0.x = bp[0];  b0.y = bp[OUT_D + 0];
    b1.x = bp[16]; b1.y = bp[OUT_D + 16];
    acc0 = __builtin_amdgcn_wmma_f32_16x16x4_f32(false, a, false, b0, (short)0, acc0, false, false);
    acc1 = __builtin_amdgcn_wmma_f32_16x16x4_f32(false, a, false, b1, (short)0, acc1, false, false);
  }
  const int mBase = mTile * 16 + ((lane >> 4) * 8);
  const int nCol  = (lane & 15);
  const float bias0 = lin_b[nCol];
  const float bias1 = lin_b[nCol + 16];
#pragma unroll
  for (int g = 0; g < 8; ++g) {
    float* orow = out + (long)(mBase + g) * OUT_D + nCol;
    orow[0]  = acc0[g] + bias0;
    orow[16] = acc1[g] + bias1;
  }
}

// ---------------------------------------------------------------------------
// Host-side launcher
// ---------------------------------------------------------------------------
extern "C" void kernel_launch(void* const* d_in, const int* in_sizes, int n_in,
                              void* d_out, int out_size, void* d_ws, size_t ws_size,
                              hipStream_t stream) {
  const float* x        = (const float*)d_in[0];
  const int*   src      = (const int*)d_in[1];
  const int*   dst      = (const int*)d_in[2];
  const float* W        = (const float*)d_in[3];
  const float* attn_l   = (const float*)d_in[4];
  const float* attn_r   = (const float*)d_in[5];
  const float* gat_bias = (const float*)d_in[6];
  const float* lin_W    = (const float*)d_in[7];
  const float* lin_b    = (const float*)d_in[8];
  float* out = (float*)d_out;

  const int  N = in_sizes[0] / D_FEAT;   // 50000
  const long E = in_sizes[1];            // 800000

  // Workspace layout (floats)
  float* ws    = (float*)d_ws;
  float* feat  = ws;                                  // N*256
  float* el    = feat  + (long)N * D_FEAT;            // N*8
  float* er    = el    + (long)N * HEADS;             // N*8
  float* emax  = er    + (long)N * HEADS;             // N*8
  float* denom = emax  + (long)N * HEADS;             // N*8
  float* ex    = denom + (long)N * HEADS;             // E*8
  float* rst   = ex    + E * HEADS;                   // N*256

  const long nSeg = (long)N * HEADS;
  const long nRst = (long)N * D_FEAT;

  // 0) init emax/-inf, denom/0, rst/0
  {
    long total = nRst;  // >= nSeg
    int blocks = (int)((total + 255) / 256);
    init_ws_kernel<<<blocks, 256, 0, stream>>>(emax, denom, rst, nSeg, nRst);
  }
  // 1) feat = x @ W  (WMMA, 16x64 strip per wave)
  {
    const int totalTiles = (N / 16) * (D_FEAT / 64);  // 3125*4 = 12500 waves
    const int blocks = (totalTiles + 7) / 8;
    gemm_feat_kernel<<<blocks, 256, 0, stream>>>(x, W, feat, N, totalTiles);
  }
  // 2) el/er
  {
    long total = (long)N * HEADS;
    int blocks = (int)((total + 255) / 256);
    node_attn_kernel<<<blocks, 256, 0, stream>>>(feat, attn_l, attn_r, el, er, total);
  }
  // 3) edge logits + segment max
  {
    long total = E * HEADS;
    int blocks = (int)((total + 255) / 256);
    edge_logit_max_kernel<<<blocks, 256, 0, stream>>>(el, er, src, dst, ex /*reuse as e buffer*/, emax, total);
  }
  // 4) exp + segment sum (in-place on ex: read e, write ex)
  {
    long total = E * HEADS;
    int blocks = (int)((total + 255) / 256);
    edge_expsum_kernel<<<blocks, 256, 0, stream>>>(ex, dst, emax, ex, denom, total);
  }
  // 5) weighted scatter-aggregate
  {
    long total = E * (D_FEAT / 4);
    int blocks = (int)((total + 255) / 256);
    aggregate_kernel<<<blocks, 256, 0, stream>>>(feat, ex, denom, src, dst, rst, total);
  }
  // 6) out = elu(rst + bias) @ lin_W + lin_b  (WMMA, 16x32 strip per wave)
  {
    const int totalTiles = N / 16;                    // 3125 waves
    const int blocks = (totalTiles + 7) / 8;
    gemm_out_kernel<<<blocks, 256, 0, stream>>>(rst, gat_bias, lin_W, lin_b, out, N, totalTiles);
  }
}